// WordRep_10557029613877
// MI455X (gfx1250) — compile-verified
//
#include <hip/hip_runtime.h>
#include <stdint.h>

// Native 16-byte vector type (accepted by __builtin_nontemporal_*)
typedef __attribute__((ext_vector_type(4))) float fv4;

// Problem constants from the reference
static constexpr int Bc = 64;
static constexpr int Lc = 512;
static constexpr int Hc = 768;
static constexpr int Wc = 255;
static constexpr int HV4 = Hc / 4; // 192 fv4 per row

// ---------------------------------------------------------------------------
// Kernel 1: per-batch exclusive prefix of word lengths -> start token index.
// starts[b][w] = 1 + sum_{j<w} len[b][j]
// One block per batch, 256 threads (8 wave32s), Hillis-Steele scan.
// ---------------------------------------------------------------------------
__global__ __launch_bounds__(256) void scan_kernel(const int* __restrict__ lengths,
                                                   int* __restrict__ starts) {
    __shared__ int sh[256];
    const int b = blockIdx.x;
    const int t = threadIdx.x;
    const int v = (t < Wc) ? lengths[b * Wc + t] : 0;
    sh[t] = v;
    __syncthreads();
#pragma unroll
    for (int off = 1; off < 256; off <<= 1) {
        int x = (t >= off) ? sh[t - off] : 0;
        __syncthreads();
        sh[t] += x;
        __syncthreads();
    }
    if (t < Wc) starts[b * Wc + t] = 1 + sh[t] - v; // 1 + exclusive prefix
}

// ---------------------------------------------------------------------------
// Kernel 2: word mean-pool + CLS copy.
// grid = B * (W+1) blocks; block = 192 threads = 6 full wave32s (EXEC all 1s,
// required for the async-to-LDS path). w == W slot handles the CLS row.
// Token rows are staged through LDS with global_load_async_to_lds_b128
// (ASYNCcnt), then reduced and scaled by 1/len. Outputs are streamed with
// non-temporal stores (write-once data; keeps the input stream resident in L2).
// ---------------------------------------------------------------------------
__global__ __launch_bounds__(192) void word_pool_kernel(const float* __restrict__ seq,
                                                        const int* __restrict__ lengths,
                                                        const int* __restrict__ starts,
                                                        float* __restrict__ out) {
    __shared__ fv4 tile[2 * HV4]; // 2 token rows x 3072 B = 6144 B LDS

    const int t  = threadIdx.x;      // 0..191
    const int bw = blockIdx.x;
    const int b  = bw / (Wc + 1);
    const int w  = bw % (Wc + 1);

    const float* baseB = seq + (size_t)b * Lc * Hc;

    if (w == Wc) {
        // cls_output[b, :] = seq[b, 0, :]  (read-once row, streamed out NT)
        const fv4 v = __builtin_nontemporal_load((const fv4*)(baseB) + t);
        __builtin_nontemporal_store(v, ((fv4*)out) + (size_t)b * HV4 + t);
        return;
    }

    const int len = lengths[b * Wc + w];
    fv4 acc = {0.f, 0.f, 0.f, 0.f};

    if (len > 0) {
        const int start = starts[b * Wc + w]; // block-uniform
        const unsigned ldsBase = (unsigned)(uintptr_t)(&tile[0]);
        const unsigned voff    = (unsigned)(t * 16); // per-lane byte offset in row

        int r = 0;
        while (r < len) {
            const int chunk = (len - r) < 2 ? (len - r) : 2;
            // Issue async global->LDS copies for up to 2 token rows.
            for (int c = 0; c < chunk; ++c) {
                const int tok = start + r + c; // block-uniform
                if (tok < Lc) {
                    const uint64_t sbase = (uint64_t)(baseB + (size_t)tok * Hc);
                    const unsigned ldsAddr = ldsBase + (unsigned)(c * Hc * 4) + voff;
                    // GVS mode: each lane copies 16 B from global address sbase plus voff to LDS address ldsAddr.
                    asm volatile("global_load_async_to_lds_b128 %0, %1, %2 offset:0"
                                 :
                                 : "v"(ldsAddr), "v"(voff), "s"(sbase)
                                 : "memory");
                }
            }
            // Drain ASYNCcnt: each lane consumes exactly the bytes it issued,
            // so no cross-lane barrier is required after the wait.
            asm volatile("s_wait_asynccnt 0" ::: "memory");
            for (int c = 0; c < chunk; ++c) {
                const int tok = start + r + c;
                if (tok < Lc) {
                    const fv4 v = tile[c * HV4 + t];
                    acc += v;
                }
            }
            r += chunk;
        }
        const float inv = 1.0f / (float)(len > 0 ? len : 1);
        acc *= inv;
    }
    // context_embedding is laid out after cls_output in d_out; streamed NT.
    fv4* outCtx = (fv4*)(out + (size_t)Bc * Hc);
    __builtin_nontemporal_store(acc, outCtx + ((size_t)b * Wc + w) * HV4 + t);
}

extern "C" void kernel_launch(void* const* d_in, const int* in_sizes, int n_in,
                              void* d_out, int out_size, void* d_ws, size_t ws_size,
                              hipStream_t stream) {
    (void)in_sizes; (void)n_in; (void)out_size; (void)ws_size;
    const float* seq     = (const float*)d_in[0]; // (B, L, H) f32
    const int*   lengths = (const int*)d_in[1];   // (B, W) i32
    float*       out     = (float*)d_out;         // [B*H cls | B*W*H ctx] f32
    int*         starts  = (int*)d_ws;            // B*W ints of scratch

    scan_kernel<<<Bc, 256, 0, stream>>>(lengths, starts);
    word_pool_kernel<<<Bc * (Wc + 1), 192, 0, stream>>>(seq, lengths, starts, out);
}